// NerfRendererPriority_23888608100543
// MI455X (gfx1250) — compile-verified
//
#include <hip/hip_runtime.h>
#include <stdint.h>

#define N_COARSE 256
#define N_FINE   128
#define RPB      8          // rays per block: one wave32 per ray
#define THREADS  256

#define ASG __attribute__((address_space(1)))
#define ASL __attribute__((address_space(3)))

typedef int v4i __attribute__((ext_vector_type(4)));

// ---- gfx1250 async global->LDS path (ASYNCcnt), probe-guarded ----
#if defined(__HIP_DEVICE_COMPILE__) && __has_builtin(__builtin_amdgcn_global_load_async_to_lds_b128)
#define HAVE_ASYNC_LDS 1
#else
#define HAVE_ASYNC_LDS 0
#if defined(__HIP_DEVICE_COMPILE__)
#warning "CDNA5 probe: __builtin_amdgcn_global_load_async_to_lds_b128 NOT available; using sync fallback"
#endif
#endif

#if defined(__HIP_DEVICE_COMPILE__) && __has_builtin(__builtin_amdgcn_s_wait_asynccnt)
#define HAVE_WAIT_ASYNC 1
#else
#define HAVE_WAIT_ASYNC 0
#endif

__device__ __forceinline__ float wave_excl_scan_mul(float p, int lane) {
    float inc = p;
#pragma unroll
    for (int off = 1; off < 32; off <<= 1) {
        float t = __shfl_up(inc, off, 32);
        if (lane >= off) inc *= t;
    }
    float exc = __shfl_up(inc, 1, 32);
    return (lane == 0) ? 1.0f : exc;
}

__device__ __forceinline__ float wave_incl_scan_add(float p, int lane) {
    float inc = p;
#pragma unroll
    for (int off = 1; off < 32; off <<= 1) {
        float t = __shfl_up(inc, off, 32);
        if (lane >= off) inc += t;
    }
    return inc;
}

__device__ __forceinline__ float wave_reduce_add(float v) {
#pragma unroll
    for (int off = 16; off; off >>= 1) v += __shfl_xor(v, off, 32);
    return v;
}

__global__ __launch_bounds__(THREADS) void nerf_render_fused(
    const float* __restrict__ sig_c,   // [N, 256]
    const float* __restrict__ sig_f,   // [N, 128]
    const float* __restrict__ rgb_f,   // [N, 128, 3]
    const float* __restrict__ bg,      // [3]
    float* __restrict__ o_img,         // [N, 3]
    float* __restrict__ o_invd,        // [N]
    float* __restrict__ o_wf,          // [N, 128]
    float* __restrict__ o_zs)          // [N, 128]
{
    __shared__ float s_cdf[RPB][N_COARSE];      // 8 KB
    __shared__ float s_sigf[RPB * N_FINE];      // 4 KB
    __shared__ float s_rgb[RPB * N_FINE * 3];   // 12 KB

    const int tid  = threadIdx.x;
    const int lane = tid & 31;
    const int wv   = tid >> 5;
    const long long ray = (long long)blockIdx.x * RPB + wv;

    const float DLOG = 0.015625f;               // coarse grid spacing 1/64
    const float FLOOR = 0.01f / 256.0f;         // ALPHA_C / S
    const float L2_10 = 3.3219280948873623f;    // log2(10)

    // ---------- split-phase prefetch of fine-pass operands into LDS ----------
    // Each thread prefetches exactly the 16 floats it will itself consume in
    // phase 3; ASYNCcnt tracks completion so the latency overlaps phase 1+2.
    {
        const float* gs = sig_f + (long long)blockIdx.x * (RPB * N_FINE) + tid * 4;
        const float* gr = rgb_f + (long long)blockIdx.x * (RPB * N_FINE * 3) + tid * 12;
#if HAVE_ASYNC_LDS
        __builtin_amdgcn_global_load_async_to_lds_b128(
            (ASG v4i*)(uintptr_t)gs,        (ASL v4i*)&s_sigf[tid * 4], 0, 0);
        __builtin_amdgcn_global_load_async_to_lds_b128(
            (ASG v4i*)(uintptr_t)(gr + 0),  (ASL v4i*)&s_rgb[tid * 12 + 0], 0, 0);
        __builtin_amdgcn_global_load_async_to_lds_b128(
            (ASG v4i*)(uintptr_t)(gr + 4),  (ASL v4i*)&s_rgb[tid * 12 + 4], 0, 0);
        __builtin_amdgcn_global_load_async_to_lds_b128(
            (ASG v4i*)(uintptr_t)(gr + 8),  (ASL v4i*)&s_rgb[tid * 12 + 8], 0, 0);
#else
        *(float4*)&s_sigf[tid * 4]     = *(const float4*)gs;
        *(float4*)&s_rgb[tid * 12 + 0] = *(const float4*)(gr + 0);
        *(float4*)&s_rgb[tid * 12 + 4] = *(const float4*)(gr + 4);
        *(float4*)&s_rgb[tid * 12 + 8] = *(const float4*)(gr + 8);
#endif
    }

    // ---------- phase 1: coarse weights -> blurred pdf -> CDF in LDS ----------
    float s8[8];
    {
        const float4* p = (const float4*)(sig_c + ray * N_COARSE + lane * 8);
        float4 A = p[0], B = p[1];
        s8[0] = A.x; s8[1] = A.y; s8[2] = A.z; s8[3] = A.w;
        s8[4] = B.x; s8[5] = B.y; s8[6] = B.z; s8[7] = B.w;
    }
    float keep[8];
    float lp = 1.0f;
#pragma unroll
    for (int i = 0; i < 8; ++i) {
        float k = __expf(-s8[i] * DLOG);
        if (lane == 31 && i == 7) k = 1.0f;   // last sample: delta = 0 -> alpha = 0
        keep[i] = k;
        lp *= k;
    }
    float T = wave_excl_scan_mul(lp, lane);    // transmittance at chunk start
    float w8[8];
#pragma unroll
    for (int i = 0; i < 8; ++i) { w8[i] = (1.0f - keep[i]) * T; T *= keep[i]; }

    // 3-tap max blur (zero-padded) + floor; seg_scale=0.5 cancels in normalization
    float wprev = __shfl_up(w8[7], 1, 32);   if (lane == 0)  wprev = 0.0f;
    float wnext = __shfl_down(w8[0], 1, 32); if (lane == 31) wnext = 0.0f;
    float W8[8], lsum = 0.0f;
#pragma unroll
    for (int i = 0; i < 8; ++i) {
        float l = i ? w8[i - 1] : wprev;
        float r = (i < 7) ? w8[i + 1] : wnext;
        float b = 0.5f * (fmaxf(l, w8[i]) + fmaxf(w8[i], r)) + FLOOR;
        W8[i] = b; lsum += b;
    }
    float isum  = wave_incl_scan_add(lsum, lane);
    float total = __shfl(isum, 31, 32);
    float rtot  = 1.0f / total;
    float run   = isum - lsum;                 // exclusive prefix
#pragma unroll
    for (int i = 0; i < 8; ++i) { run += W8[i]; s_cdf[wv][lane * 8 + i] = run * rtot; }

    // ---------- phase 2: inverse-CDF sampling (4 samples per lane) ----------
    float zlf[4];
#pragma unroll
    for (int t = 0; t < 4; ++t) {
        int j = lane * 4 + t;
        float u = ((float)j + 0.5f) * (1.0f / 128.0f);
        int lo = 0, hi = N_COARSE;
        while (lo < hi) {                      // searchsorted(side='right')
            int mid = (lo + hi) >> 1;
            if (s_cdf[wv][mid] <= u) lo = mid + 1; else hi = mid;
        }
        int below = lo - 1; if (below < 0) below = 0; if (below > 255) below = 255;
        int above = lo;     if (above > 255) above = 255;
        float cb = s_cdf[wv][below];
        float ca = s_cdf[wv][above];
        float den = ca - cb; den = (den < 1e-5f) ? 1.0f : den;
        float tt = (u - cb) / den;
        float bb = -2.0f + (float)below * DLOG;
        float ba = -2.0f + (float)above * DLOG;
        zlf[t] = bb + tt * (ba - bb);
    }

    // ---------- phase 3: fine compositing ----------
    float znext0 = __shfl_down(zlf[0], 1, 32);
    float d0 = zlf[1] - zlf[0];
    float d1 = zlf[2] - zlf[1];
    float d2 = zlf[3] - zlf[2];
    float d3 = (lane == 31) ? 0.0f : (znext0 - zlf[3]);
    float dd[4] = {d0, d1, d2, d3};

#if HAVE_WAIT_ASYNC
    __builtin_amdgcn_s_wait_asynccnt(0);       // s_wait_asynccnt 0
#elif HAVE_ASYNC_LDS
    asm volatile("s_wait_asynccnt 0" ::: "memory");
#endif

    float4 sf = *(const float4*)&s_sigf[wv * N_FINE + lane * 4];
    float fs[4] = {sf.x, sf.y, sf.z, sf.w};
    float fkeep[4]; float flp = 1.0f;
#pragma unroll
    for (int i = 0; i < 4; ++i) { float k = __expf(-fs[i] * dd[i]); fkeep[i] = k; flp *= k; }
    float Tf = wave_excl_scan_mul(flp, lane);
    float wf[4];
#pragma unroll
    for (int i = 0; i < 4; ++i) { wf[i] = (1.0f - fkeep[i]) * Tf; Tf *= fkeep[i]; }

    float4 c0 = *(const float4*)&s_rgb[wv * N_FINE * 3 + lane * 12 + 0];
    float4 c1 = *(const float4*)&s_rgb[wv * N_FINE * 3 + lane * 12 + 4];
    float4 c2 = *(const float4*)&s_rgb[wv * N_FINE * 3 + lane * 12 + 8];

    float ws = wf[0] + wf[1] + wf[2] + wf[3];
    float ir = wf[0] * c0.x + wf[1] * c0.w + wf[2] * c1.z + wf[3] * c2.y;
    float ig = wf[0] * c0.y + wf[1] * c1.x + wf[2] * c1.w + wf[3] * c2.z;
    float ib = wf[0] * c0.z + wf[1] * c1.y + wf[2] * c2.x + wf[3] * c2.w;
    float id = wf[0] * exp2f(-zlf[0] * L2_10) + wf[1] * exp2f(-zlf[1] * L2_10)
             + wf[2] * exp2f(-zlf[2] * L2_10) + wf[3] * exp2f(-zlf[3] * L2_10);

    ws = wave_reduce_add(ws);
    ir = wave_reduce_add(ir);
    ig = wave_reduce_add(ig);
    ib = wave_reduce_add(ib);
    id = wave_reduce_add(id);

    // ---------- outputs ----------
    long long ob = ray * N_FINE + lane * 4;
    *(float4*)&o_wf[ob] = make_float4(wf[0], wf[1], wf[2], wf[3]);
    *(float4*)&o_zs[ob] = make_float4((zlf[0] + 2.0f) * 0.25f, (zlf[1] + 2.0f) * 0.25f,
                                      (zlf[2] + 2.0f) * 0.25f, (zlf[3] + 2.0f) * 0.25f);
    if (lane == 0) {
        float rem = 1.0f - ws;
        o_img[ray * 3 + 0] = ir + rem * bg[0];
        o_img[ray * 3 + 1] = ig + rem * bg[1];
        o_img[ray * 3 + 2] = ib + rem * bg[2];
        o_invd[ray] = id;
    }
}

extern "C" void kernel_launch(void* const* d_in, const int* in_sizes, int n_in,
                              void* d_out, int out_size, void* d_ws, size_t ws_size,
                              hipStream_t stream) {
    const float* sig_c = (const float*)d_in[0];
    const float* sig_f = (const float*)d_in[1];
    const float* rgb_f = (const float*)d_in[2];
    const float* bg    = (const float*)d_in[3];
    const long long n_rays = (long long)in_sizes[0] / N_COARSE;   // 65536

    // d_out layout (flat concat in return order):
    // image [N,3] | invdepth [N] | w_fine [N,128] | z_log_s [N,128]
    float* out    = (float*)d_out;
    float* o_img  = out;
    float* o_invd = out + n_rays * 3;
    float* o_wf   = out + n_rays * 4;
    float* o_zs   = out + n_rays * (4 + N_FINE);

    dim3 grid((unsigned)(n_rays / RPB));
    dim3 block(THREADS);
    hipLaunchKernelGGL(nerf_render_fused, grid, block, 0, stream,
                       sig_c, sig_f, rgb_f, bg, o_img, o_invd, o_wf, o_zs);
}